// AffineQuantizedMSEObserver_86260123173369
// MI455X (gfx1250) — compile-verified
//
#include <hip/hip_runtime.h>
#include <hip/hip_bf16.h>
#include <stdint.h>

// AffineQuantizedMSEObserver for MI455X (gfx1250, wave32).
// One 256-thread block (8 waves) per row. Row (32KB) staged into LDS with
// CDNA5 async global->LDS loads, pulled into registers once; the 100-step
// threshold sweep runs out of VGPRs, batched 10 steps at a time so the
// squared-error FMAs form 10 independent accumulation chains (ILP) and
// block barriers drop from 200 to 20 per row. Scalar inner loop with the
// clamp forced to a single v_med3_f32 (5 VALU ops per element-step); the
// lane-0 LDS store is hoisted out of the per-step loop so the epilogue pays
// one EXEC transition per batch instead of ten.

#define ROW_K   8192
#define STEPS   100
#define GBATCH  10              // steps per batch (independent acc chains)
#define TPB     256
#define NWAVES  (TPB / 32)      // 8
#define EPT     (ROW_K / TPB)   // 32 floats per thread
#define V4PT    (EPT / 4)       // 8 float4 per thread
#define QEPS    1.1920928955078125e-07f

static_assert(STEPS % GBATCH == 0, "batch must divide steps");

// One element-step: exactly 5 VALU ops.
//   t = x * inv_s          -> v_mul_f32
//   q = rndne(t)           -> v_rndne_f32
//   q = med3(q,-128,127)   -> v_med3_f32
//   e = fma(q, s, -x)      -> v_fma_f32
//   acc = fma(e, e, acc)   -> v_fma_f32
__device__ __forceinline__ float qerr_acc(float xv, float s, float inv_s, float acc) {
    float q = rintf(xv * inv_s);
    q = __builtin_amdgcn_fmed3f(q, -128.0f, 127.0f);
    const float e = __builtin_fmaf(q, s, -xv);
    return __builtin_fmaf(e, e, acc);
}

__global__ __launch_bounds__(TPB) void observer_kernel(const float* __restrict__ x,
                                                       float* __restrict__ out,
                                                       int nrows) {
    __shared__ __align__(16) float tile[ROW_K];     // 32 KB row staging
    __shared__ float red[NWAVES * GBATCH];          // 8 waves x 10 step-partials
    __shared__ float rmn[NWAVES];
    __shared__ float rmx[NWAVES];

    const int row = blockIdx.x;
    const int tid = threadIdx.x;
    const float* rowp = x + (size_t)row * ROW_K;

    // ---- Stage whole row into LDS via CDNA5 async loads (ASYNCcnt path) ----
    // GVS mode: mem = SADDR(64b uniform row base) + VADDR(32b byte offset);
    // VDST vgpr carries the LDS byte address.
    const uint32_t lds_base = (uint32_t)(uintptr_t)(&tile[0]);
    #pragma unroll
    for (int j = 0; j < V4PT; ++j) {
        uint32_t byte_off = (uint32_t)(tid * 16 + j * (TPB * 16));
        uint32_t lds_addr = lds_base + byte_off;
        asm volatile("global_load_async_to_lds_b128 %0, %1, %2"
                     :
                     : "v"(lds_addr), "v"(byte_off), "s"(rowp)
                     : "memory");
    }
#if __has_builtin(__builtin_amdgcn_s_wait_asynccnt)
    __builtin_amdgcn_s_wait_asynccnt(0);
#else
    asm volatile("s_wait_asynccnt 0x0" ::: "memory");
#endif
    __syncthreads();

    // ---- Pull this thread's 32 elements into registers (ds_load_b128 x8) ----
    float4 v[V4PT];
    #pragma unroll
    for (int j = 0; j < V4PT; ++j) {
        v[j] = *(const float4*)&tile[tid * 4 + j * (TPB * 4)];
    }

    // ---- Per-row min / max -> range ----
    float mn = v[0].x, mx = v[0].x;
    #pragma unroll
    for (int j = 0; j < V4PT; ++j) {
        mn = fminf(mn, fminf(fminf(v[j].x, v[j].y), fminf(v[j].z, v[j].w)));
        mx = fmaxf(mx, fmaxf(fmaxf(v[j].x, v[j].y), fmaxf(v[j].z, v[j].w)));
    }
    #pragma unroll
    for (int off = 16; off > 0; off >>= 1) {
        mn = fminf(mn, __shfl_xor(mn, off, 32));
        mx = fmaxf(mx, __shfl_xor(mx, off, 32));
    }
    if ((tid & 31) == 0) { rmn[tid >> 5] = mn; rmx[tid >> 5] = mx; }
    __syncthreads();
    float gmn = rmn[0], gmx = rmx[0];
    #pragma unroll
    for (int w = 1; w < NWAVES; ++w) {
        gmn = fminf(gmn, rmn[w]);
        gmx = fmaxf(gmx, rmx[w]);
    }
    const float range  = fmaxf(fabsf(gmn), gmx);
    const float step_t = range * (1.0f / (float)STEPS);

    // ---- 100-step sweep, batched GBATCH steps for ILP + fewer barriers ----
    float opt    = 1.0e9f;   // tracked by thread 0
    float best_t = 0.0f;

    #pragma unroll 1
    for (int base = 0; base < STEPS; base += GBATCH) {
        float sg[GBATCH], isg[GBATCH], acc[GBATCH];
        #pragma unroll
        for (int g = 0; g < GBATCH; ++g) {
            const float thres = step_t * (float)(base + g + 1);
            sg[g]  = fmaxf(thres * (1.0f / 127.5f), QEPS);
            isg[g] = 1.0f / sg[g];
            acc[g] = 0.0f;
        }

        // Element loop outer, step loop innermost: consecutive FMAs hit
        // different acc[g] -> 10 independent dependency chains.
        #pragma unroll 2
        for (int j = 0; j < V4PT; ++j) {
            #pragma unroll
            for (int c = 0; c < 4; ++c) {
                const float xv = (c == 0) ? v[j].x : (c == 1) ? v[j].y
                               : (c == 2) ? v[j].z : v[j].w;
                #pragma unroll
                for (int g = 0; g < GBATCH; ++g) {
                    acc[g] = qerr_acc(xv, sg[g], isg[g], acc[g]);
                }
            }
        }

        // Wave butterfly per step (all lanes), then ONE lane-0 store block
        // (single EXEC transition), then one cross-wave pass by thread 0.
        #pragma unroll
        for (int g = 0; g < GBATCH; ++g) {
            #pragma unroll
            for (int off = 16; off > 0; off >>= 1) acc[g] += __shfl_xor(acc[g], off, 32);
        }
        if ((tid & 31) == 0) {
            #pragma unroll
            for (int g = 0; g < GBATCH; ++g) red[(tid >> 5) * GBATCH + g] = acc[g];
        }
        __syncthreads();
        if (tid == 0) {
            #pragma unroll
            for (int g = 0; g < GBATCH; ++g) {
                float loss = 0.0f;
                #pragma unroll
                for (int w = 0; w < NWAVES; ++w) loss += red[w * GBATCH + g];
                loss *= (1.0f / (float)ROW_K);
                const float thres = step_t * (float)(base + g + 1);
                if (loss < opt) { opt = loss; best_t = thres; }  // in step order
            }
        }
        __syncthreads();   // protect red[] reuse next batch
    }

    if (tid == 0) {
        // Finite data always beats the 1e9 sentinel; keep reference fallback.
        const bool hit = (opt < 1.0e9f);
        out[row]         = hit ? -best_t : gmn;
        out[nrows + row] = hit ?  best_t : gmx;
    }
}

extern "C" void kernel_launch(void* const* d_in, const int* in_sizes, int n_in,
                              void* d_out, int out_size, void* d_ws, size_t ws_size,
                              hipStream_t stream) {
    (void)n_in; (void)d_ws; (void)ws_size; (void)in_sizes;
    const float* x = (const float*)d_in[0];
    float* out = (float*)d_out;
    const int nrows = out_size / 2;              // 8192
    observer_kernel<<<dim3(nrows), dim3(TPB), 0, stream>>>(x, out, nrows);
}